// PDHG_87660282511589
// MI455X (gfx1250) — compile-verified
//
#include <hip/hip_runtime.h>
#include <math.h>
#include <stdint.h>

// Problem constants from the reference: (P, C, NX, NY, NT) = (2, 1, 128, 128, 8), T = 128
#define PP   2
#define NXX  128
#define NYY  128
#define NTT  8
#define NSITE (PP * NXX * NYY * NTT)   // 262144 floats per field
#define TILE 16
#define HALO 18                         // TILE + 2 (one-site halo each side)
#define TSTEPS 128

__device__ __forceinline__ int sbase(int p, int x, int y) {
    return ((p * NXX + x) * NYY + y) * NTT;
}
__device__ __forceinline__ int qbase(int p, int d, int x, int y) {
    return (((p * 3 + d) * NXX + x) * NYY + y) * NTT;
}

__device__ __forceinline__ void load8(float* r, const float* g) {
    float4 a = reinterpret_cast<const float4*>(g)[0];
    float4 b = reinterpret_cast<const float4*>(g)[1];
    r[0] = a.x; r[1] = a.y; r[2] = a.z; r[3] = a.w;
    r[4] = b.x; r[5] = b.y; r[6] = b.z; r[7] = b.w;
}
__device__ __forceinline__ void store8(float* g, const float* r) {
    float4 a = make_float4(r[0], r[1], r[2], r[3]);
    float4 b = make_float4(r[4], r[5], r[6], r[7]);
    reinterpret_cast<float4*>(g)[0] = a;
    reinterpret_cast<float4*>(g)[1] = b;
}

// CDNA5 async global->LDS copy (16B per enabled lane), tracked by ASYNCcnt.
__device__ __forceinline__ void async_copy_b128(uint32_t lds_byte_addr, const float* gsrc) {
    asm volatile("global_load_async_to_lds_b128 %0, %1, off"
                 :: "v"(lds_byte_addr), "v"(gsrc)
                 : "memory");
}
__device__ __forceinline__ void wait_async0() {
    asm volatile("s_wait_asynccnt 0" ::: "memory");
}

__global__ void pdhg_init_kernel(const float* __restrict__ x5,
                                 float* __restrict__ x,
                                 float* __restrict__ p,
                                 float* __restrict__ xbar0,
                                 float* __restrict__ q0) {
    int i = blockIdx.x * blockDim.x + threadIdx.x;      // grid covers NSITE
    float v = x5[i];
    x[i] = v;
    p[i] = v;
    xbar0[i] = v;
    q0[i] = 0.0f;
    q0[i + NSITE] = 0.0f;
    q0[i + 2 * NSITE] = 0.0f;
}

__global__ void __launch_bounds__(256)
pdhg_step_kernel(const float* __restrict__ xnoisy,
                 const float* __restrict__ lam,
                 const float* __restrict__ s_tau,
                 const float* __restrict__ s_sigma,
                 const float* __restrict__ s_theta,
                 const float* __restrict__ xbar_r,
                 float* __restrict__ xbar_w,
                 const float* __restrict__ q_r,
                 float* __restrict__ q_w,
                 float* __restrict__ p,
                 float* __restrict__ x) {
    __shared__ float sx[HALO * HALO * NTT];             // 18*18*8 f32 = 10.1 KB

    const int b    = blockIdx.x;                        // PP * 64 blocks
    const int pp   = b >> 6;
    const int tile = b & 63;
    const int tx0  = (tile >> 3) * TILE;
    const int ty0  = (tile & 7) * TILE;
    const int tid  = threadIdx.x;                       // 256 threads = 8 wave32

    // --- Stage xbar tile + halo into LDS via async copies (two B128 per site) ---
    for (int h = tid; h < HALO * HALO; h += 256) {
        int hx = (tx0 - 1 + (h / HALO) + NXX) & (NXX - 1);   // circular roll
        int hy = (ty0 - 1 + (h % HALO) + NYY) & (NYY - 1);
        const float* g = xbar_r + sbase(pp, hx, hy);
        uint32_t lo = (uint32_t)(uintptr_t)(&sx[h * NTT]);   // low 32 bits = LDS offset
        async_copy_b128(lo, g);
        async_copy_b128(lo + 16u, g + 4);
    }
    wait_async0();
    __syncthreads();

    // --- Step scalars (uniform loads; sigmoid on device) ---
    const float rL = 0.27735009811261456f;              // 1/sqrt(13)
    const float sg = (1.0f / (1.0f + __expf(-s_sigma[0]))) * rL;
    const float tu = (1.0f / (1.0f + __expf(-s_tau[0]))) * rL;
    const float th = 1.0f / (1.0f + __expf(-s_theta[0]));

    const int lx = tid >> 4, ly = tid & 15;
    const int gx = tx0 + lx, gy = ty0 + ly;
    const int xm = (gx - 1 + NXX) & (NXX - 1);
    const int ym = (gy - 1 + NYY) & (NYY - 1);

    const float* sc = &sx[((lx + 1) * HALO + (ly + 1)) * NTT];
    float xbc[8], xbxp[8], xbxm[8], xbyp[8], xbym[8];
    load8(xbc,  sc);
    load8(xbxp, sc + HALO * NTT);
    load8(xbxm, sc - HALO * NTT);
    load8(xbyp, sc + NTT);
    load8(xbym, sc - NTT);

    const int ib = sbase(pp, gx, gy);

    float xn[8], pv[8], la[8];
    load8(xn, xnoisy + ib);
    load8(pv, p + ib);
    load8(la, lam + ib);

    // p' = (p + sig*(xbar - xnoisy)) / (1 + sig)
    const float rs = 1.0f / (1.0f + sg);
    float pn[8];
#pragma unroll
    for (int t = 0; t < 8; ++t) pn[t] = (pv[t] + sg * (xbc[t] - xn[t])) * rs;
    store8(p + ib, pn);

    float gh[8];

    // d = 0 (x axis): own q0' plus recomputed q0' at (x-1) for the divergence
    {
        float q0[8], q0m[8], lm[8], q0n[8];
        load8(q0,  q_r + qbase(pp, 0, gx, gy));
        load8(q0m, q_r + qbase(pp, 0, xm, gy));
        load8(lm,  lam + sbase(pp, xm, gy));
#pragma unroll
        for (int t = 0; t < 8; ++t) {
            float v = q0[t] + sg * (xbxp[t] - xbc[t]);
            q0n[t] = v * la[t] / fmaxf(la[t], fabsf(v));
            float w = q0m[t] + sg * (xbc[t] - xbxm[t]);
            float q0nm = w * lm[t] / fmaxf(lm[t], fabsf(w));
            gh[t] = q0nm - q0n[t];
        }
        store8(q_w + qbase(pp, 0, gx, gy), q0n);
    }
    // d = 1 (y axis)
    {
        float q1[8], q1m[8], lm[8], q1n[8];
        load8(q1,  q_r + qbase(pp, 1, gx, gy));
        load8(q1m, q_r + qbase(pp, 1, gx, ym));
        load8(lm,  lam + sbase(pp, gx, ym));
#pragma unroll
        for (int t = 0; t < 8; ++t) {
            float v = q1[t] + sg * (xbyp[t] - xbc[t]);
            q1n[t] = v * la[t] / fmaxf(la[t], fabsf(v));
            float w = q1m[t] + sg * (xbc[t] - xbym[t]);
            float q1nm = w * lm[t] / fmaxf(lm[t], fabsf(w));
            gh[t] += q1nm - q1n[t];
        }
        store8(q_w + qbase(pp, 1, gx, gy), q1n);
    }
    // d = 2 (t axis, fully in-register: NT = 8 per lane)
    {
        float q2[8], q2n[8];
        load8(q2, q_r + qbase(pp, 2, gx, gy));
#pragma unroll
        for (int t = 0; t < 8; ++t) {
            float v = q2[t] + sg * (xbc[(t + 1) & 7] - xbc[t]);
            q2n[t] = v * la[t] / fmaxf(la[t], fabsf(v));
        }
#pragma unroll
        for (int t = 0; t < 8; ++t) gh[t] += q2n[(t - 1) & 7] - q2n[t];
        store8(q_w + qbase(pp, 2, gx, gy), q2n);
    }

    // x1 = x0 - tau*(p' + div);  xbar' = x1 + th*(x1 - x0)
    float x0v[8], x1[8], xbn[8];
    load8(x0v, x + ib);
#pragma unroll
    for (int t = 0; t < 8; ++t) {
        x1[t]  = x0v[t] - tu * (pn[t] + gh[t]);
        xbn[t] = x1[t] + th * (x1[t] - x0v[t]);
    }
    store8(x + ib, x1);                 // in-place: becomes x0 next step; final -> d_out
    store8(xbar_w + ib, xbn);
}

extern "C" void kernel_launch(void* const* d_in, const int* in_sizes, int n_in,
                              void* d_out, int out_size, void* d_ws, size_t ws_size,
                              hipStream_t stream) {
    (void)in_sizes; (void)n_in; (void)out_size; (void)ws_size;

    const float* x5      = (const float*)d_in[0];   // x_5D
    const float* lam     = (const float*)d_in[1];   // lambda_reg
    const float* s_tau   = (const float*)d_in[2];
    const float* s_sigma = (const float*)d_in[3];
    const float* s_theta = (const float*)d_in[4];
    // d_in[5] = T (reference constant 128; used at compile time)

    float* x  = (float*)d_out;                      // x0/x1 buffer, in-place
    float* ws = (float*)d_ws;                       // needs 9*NSITE floats ~ 9.4 MB
    float* xbar[2] = { ws,             ws + NSITE };
    float* q[2]    = { ws + 2 * NSITE, ws + 5 * NSITE };   // 3*NSITE each
    float* p       = ws + 8 * NSITE;

    pdhg_init_kernel<<<NSITE / 256, 256, 0, stream>>>(x5, x, p, xbar[0], q[0]);

    for (int s = 0; s < TSTEPS; ++s) {
        const int r = s & 1, w = r ^ 1;
        pdhg_step_kernel<<<PP * 64, 256, 0, stream>>>(
            x5, lam, s_tau, s_sigma, s_theta,
            xbar[r], xbar[w], q[r], q[w], p, x);
    }
}